// GraphConvBinaryClassifier_74947179315796
// MI455X (gfx1250) — compile-verified
//
#include <hip/hip_runtime.h>
#include <hip/hip_bf16.h>
#include <math.h>

#define N_NODES 50000
#define N_EDGES 800000
#define IN_DIM  128
#define HDIM    64

typedef float v2f __attribute__((ext_vector_type(2)));
typedef float v8f __attribute__((ext_vector_type(8)));

// ---------------- utility kernels ----------------

__global__ void zero_kernel(float* __restrict__ p, int n) {
    int i = blockIdx.x * blockDim.x + threadIdx.x;
    if (i < n) p[i] = 0.0f;
}

__global__ void degree_kernel(const int* __restrict__ src, const int* __restrict__ dst,
                              float* __restrict__ deg_out, float* __restrict__ deg_in, int nE) {
    int e = blockIdx.x * blockDim.x + threadIdx.x;
    if (e < nE) {
        unsafeAtomicAdd(&deg_out[src[e]], 1.0f);
        unsafeAtomicAdd(&deg_in[dst[e]], 1.0f);
    }
}

__global__ void norm_inplace_kernel(float* __restrict__ p, int n) {
    int i = blockIdx.x * blockDim.x + threadIdx.x;
    if (i < n) p[i] = rsqrtf(fmaxf(p[i], 1.0f));
}

// ---------------- dense GEMM via fp32 WMMA ----------------
// y[N x 64] = (x * norm_src[:,None]) @ W   with W [K x 64] row-major.
// One wave per 16x16 output tile; 4 waves/block -> 16 rows x 64 cols per block.
// N_NODES = 3125*16 exactly, so no bounds checks needed.
template <int K>
__global__ void gemm_norm_kernel(const float* __restrict__ x,
                                 const float* __restrict__ norm_src,
                                 const float* __restrict__ W,
                                 float* __restrict__ y) {
    const int wave = threadIdx.x >> 5;     // 0..3 -> column tile
    const int lane = threadIdx.x & 31;
    const int half = lane >> 4;            // 0: lanes 0-15, 1: lanes 16-31
    const int l    = lane & 15;
    const int m0   = blockIdx.x * 16;
    const int row  = m0 + l;               // A-matrix row for this lane
    const int col  = wave * 16 + l;        // B/C/D column for this lane

    const float nrm = norm_src[row];
    v8f acc = {};

    #pragma unroll 4
    for (int k0 = 0; k0 < K; k0 += 4) {
        const int ka = k0 + half * 2;      // ISA A layout: VGPR0/1 = K {ka, ka+1}
        // A fragment: 2 consecutive K elements of one row (8B aligned: K%2==0)
        v2f a;
        const float2 xa = *(const float2*)(x + row * K + ka);
        a.x = xa.x * nrm;
        a.y = xa.y * nrm;
        // B fragment: W rows ka, ka+1 at column `col`
        v2f b;
        b.x = W[ka * 64 + col];
        b.y = W[(ka + 1) * 64 + col];
        acc = __builtin_amdgcn_wmma_f32_16x16x4_f32(
            /*neg_a=*/false, a, /*neg_b=*/false, b,
            /*c_mod=*/(short)0, acc, /*reuse_a=*/false, /*reuse_b=*/false);
    }

    // C/D layout: VGPR r -> row m0 + r + half*8, column `col`
    #pragma unroll
    for (int r = 0; r < 8; ++r) {
        y[(m0 + r + half * 8) * 64 + col] = acc[r];
    }
}

// ---------------- SpMM: agg[dst] += y[src] (64 features/edge) ----------------
__global__ void spmm_kernel(const float* __restrict__ y,
                            const int* __restrict__ src, const int* __restrict__ dst,
                            float* __restrict__ agg, int nE) {
    long long t = (long long)blockIdx.x * blockDim.x + threadIdx.x;
    int e = (int)(t >> 6);
    int f = (int)(t & 63);
    if (e < nE) {
        int s = src[e], d = dst[e];
        unsafeAtomicAdd(&agg[(long long)d * 64 + f], y[(long long)s * 64 + f]);
    }
}

// ---------------- xnext = relu(agg * norm_dst + bias) ----------------
__global__ void finalize_kernel(const float* __restrict__ agg,
                                const float* __restrict__ norm_dst,
                                const float* __restrict__ bias,
                                float* __restrict__ xnext, int n) {
    int t = blockIdx.x * blockDim.x + threadIdx.x;
    if (t < n) {
        int i = t >> 6, f = t & 63;
        xnext[t] = fmaxf(agg[t] * norm_dst[i] + bias[f], 0.0f);
    }
}

// ---------------- mean pool partial sums ----------------
__global__ void pool_kernel(const float* __restrict__ x, float* __restrict__ pool) {
    int f = threadIdx.x;  // 64 threads
    float s = 0.0f;
    for (int i = blockIdx.x; i < N_NODES; i += gridDim.x)
        s += x[(long long)i * 64 + f];
    unsafeAtomicAdd(&pool[f], s);
}

// ---------------- MLP head + sigmoid (single tiny block) ----------------
__global__ void head_kernel(const float* __restrict__ pool,
                            const float* __restrict__ Wf1, const float* __restrict__ bf1,
                            const float* __restrict__ Wf2, const float* __restrict__ bf2,
                            float* __restrict__ out) {
    __shared__ float hg[64];
    __shared__ float z1[32];
    int t = threadIdx.x;
    hg[t] = pool[t] * (1.0f / (float)N_NODES);
    __syncthreads();
    if (t < 32) {
        float s = bf1[t];
        #pragma unroll 8
        for (int f = 0; f < 64; ++f) s += hg[f] * Wf1[f * 32 + t];
        z1[t] = fmaxf(s, 0.0f);
    }
    __syncthreads();
    if (t == 0) {
        float s = bf2[0];
        #pragma unroll 8
        for (int j = 0; j < 32; ++j) s += z1[j] * Wf2[j];
        out[0] = 1.0f / (1.0f + expf(-s));
    }
}

// ---------------- launch ----------------

extern "C" void kernel_launch(void* const* d_in, const int* in_sizes, int n_in,
                              void* d_out, int out_size, void* d_ws, size_t ws_size,
                              hipStream_t stream) {
    const float* h   = (const float*)d_in[0];
    const float* W1  = (const float*)d_in[1];
    const float* b1  = (const float*)d_in[2];
    const float* W2  = (const float*)d_in[3];
    const float* b2  = (const float*)d_in[4];
    const float* W3  = (const float*)d_in[5];
    const float* b3  = (const float*)d_in[6];
    const float* Wf1 = (const float*)d_in[7];
    const float* bf1 = (const float*)d_in[8];
    const float* Wf2 = (const float*)d_in[9];
    const float* bf2 = (const float*)d_in[10];
    const int*   ei  = (const int*)d_in[11];   // [2, N_EDGES] int32
    const int* src = ei;
    const int* dst = ei + N_EDGES;
    float* out = (float*)d_out;

    // workspace layout (floats)
    float* ws       = (float*)d_ws;
    float* norm_src = ws;                         // 50000 (deg_out -> rsqrt)
    float* norm_dst = norm_src + N_NODES;         // 50000 (deg_in  -> rsqrt)
    float* bufA     = norm_dst + N_NODES;         // 50000*64
    float* bufB     = bufA + (size_t)N_NODES * 64;// 50000*64
    float* pool     = bufB + (size_t)N_NODES * 64;// 64

    const int T = 256;
    const int nFeatElems = N_NODES * HDIM;                 // 3.2M
    const int gFeat = (nFeatElems + T - 1) / T;
    const int gSpmm = (int)(((long long)N_EDGES * 64 + T - 1) / T);
    const int gGemm = N_NODES / 16;                        // 3125

    // 1) degrees -> normalization factors
    zero_kernel<<<(2 * N_NODES + T - 1) / T, T, 0, stream>>>(norm_src, 2 * N_NODES);
    degree_kernel<<<(N_EDGES + T - 1) / T, T, 0, stream>>>(src, dst, norm_src, norm_dst, N_EDGES);
    norm_inplace_kernel<<<(2 * N_NODES + T - 1) / T, T, 0, stream>>>(norm_src, 2 * N_NODES);

    // 2) layer 1: (h*norm_src)@W1 -> SpMM -> relu(.*norm_dst + b1)   [x1 in bufA]
    gemm_norm_kernel<IN_DIM><<<gGemm, 128, 0, stream>>>(h, norm_src, W1, bufA);
    zero_kernel<<<gFeat, T, 0, stream>>>(bufB, nFeatElems);
    spmm_kernel<<<gSpmm, T, 0, stream>>>(bufA, src, dst, bufB, N_EDGES);
    finalize_kernel<<<gFeat, T, 0, stream>>>(bufB, norm_dst, b1, bufA, nFeatElems);

    // 3) layer 2  [x2 in bufB]
    gemm_norm_kernel<HDIM><<<gGemm, 128, 0, stream>>>(bufA, norm_src, W2, bufB);
    zero_kernel<<<gFeat, T, 0, stream>>>(bufA, nFeatElems);
    spmm_kernel<<<gSpmm, T, 0, stream>>>(bufB, src, dst, bufA, N_EDGES);
    finalize_kernel<<<gFeat, T, 0, stream>>>(bufA, norm_dst, b2, bufB, nFeatElems);

    // 4) layer 3  [x3 in bufA]
    gemm_norm_kernel<HDIM><<<gGemm, 128, 0, stream>>>(bufB, norm_src, W3, bufA);
    zero_kernel<<<gFeat, T, 0, stream>>>(bufB, nFeatElems);
    spmm_kernel<<<gSpmm, T, 0, stream>>>(bufA, src, dst, bufB, N_EDGES);
    finalize_kernel<<<gFeat, T, 0, stream>>>(bufB, norm_dst, b3, bufA, nFeatElems);

    // 5) mean pool + MLP head + sigmoid
    zero_kernel<<<1, 64, 0, stream>>>(pool, 64);
    pool_kernel<<<256, 64, 0, stream>>>(bufA, pool);
    head_kernel<<<1, 64, 0, stream>>>(pool, Wf1, bf1, Wf2, bf2, out);
}